// FacebookAdaptiveSoftmax_70360154243551
// MI455X (gfx1250) — compile-verified
//
#include <hip/hip_runtime.h>
#include <stdint.h>

// ---------------------------------------------------------------------------
// FacebookAdaptiveSoftmax forward for MI455X (gfx1250, wave32).
// Five GEMMs of the shared form  C[M,Nv] = A[M,K] * B[Nv,K]^T  (both row-major)
// computed with V_WMMA_F32_16X16X4_F32 (native fp32 matrix op -> reference
// precision). K-tiles are streamed into double-buffered LDS with
// GLOBAL_LOAD_ASYNC_TO_LDS_B128 (ASYNCcnt-tracked, no VGPR round-trip),
// epilogue uses non-temporal stores (824 MB streaming output must not evict
// the L2-resident weights) with the bucket row-mask fused in.
// ---------------------------------------------------------------------------

typedef __attribute__((ext_vector_type(2))) float v2f;
typedef __attribute__((ext_vector_type(8))) float v8f;
typedef int v4i __attribute__((vector_size(4 * sizeof(int))));
typedef __attribute__((address_space(1))) v4i* gv4i_p;   // global int4*
typedef __attribute__((address_space(3))) v4i* lv4i_p;   // LDS int4*

#define BM 128
#define BN 128
#define BK 32
#define LDSK 36   // padded LDS row stride (floats); multiple of 4, spreads banks

#define I32_MIN (-2147483647 - 1)
#define I32_MAX 2147483647

#if __has_builtin(__builtin_amdgcn_global_load_async_to_lds_b128)
#define HAVE_ASYNC_LDS 1
#else
#define HAVE_ASYNC_LDS 0
#endif

#if __has_builtin(__builtin_amdgcn_s_wait_asynccnt)
#define WAIT_ASYNC(n) __builtin_amdgcn_s_wait_asynccnt(n)
#else
#define WAIT_ASYNC(n) asm volatile("s_wait_asynccnt %0" ::"n"(n) : "memory")
#endif

__device__ __forceinline__ void async_cp16(const float* g, float* l)
{
#if HAVE_ASYNC_LDS
    __builtin_amdgcn_global_load_async_to_lds_b128(
        (gv4i_p)(uintptr_t)g,   // global: flat VA == AS1 VA, inttoptr is safe
        (lv4i_p)l,              // LDS: C-style addrspacecast strips aperture
        /*imm offset=*/0, /*cpol=*/0);
#else
    (void)g; (void)l;
#endif
}

// Issue one BM x BK (A) + BN x BK (B) tile: 8 async b128 ops per thread.
__device__ __forceinline__ void issue_tile(
    const float* __restrict__ A, const float* __restrict__ B,
    float* lA, float* lB, int blockM, int blockN, int Nv, int K, int kt, int tid)
{
#pragma unroll
    for (int i = 0; i < 4; ++i) {
        int slot = tid + i * 256;          // 0..1023 = 128 rows x 8 float4 cols
        int r    = slot >> 3;
        int c4   = slot & 7;
        async_cp16(A + (size_t)(blockM + r) * K + kt + c4 * 4,
                   &lA[r * LDSK + c4 * 4]);
    }
#pragma unroll
    for (int i = 0; i < 4; ++i) {
        int slot = tid + i * 256;
        int r    = slot >> 3;
        int c4   = slot & 7;
        int nb   = blockN + r;
        if (nb >= Nv) nb = Nv - 1;  // clamp: dup row only feeds cols >= Nv,
                                    // which the epilogue never stores
        async_cp16(B + (size_t)nb * K + kt + c4 * 4,
                   &lB[r * LDSK + c4 * 4]);
    }
}

// One block: 256 threads = 8 waves. Waves tiled 4 (M) x 2 (N); each wave owns a
// 32x64 sub-tile = 2x4 WMMA 16x16 accumulators.
__global__ __launch_bounds__(256) void adasoft_wmma_gemm_nt(
    const float* __restrict__ A,   // [M, K] row-major
    const float* __restrict__ B,   // [Nv, K] row-major (i.e. B^T of the GEMM)
    float* __restrict__ C,         // out rows stride ldc, columns at col_off
    const int* __restrict__ target,
    int M, int Nv, int K, int ldc, int col_off, int lo, int hi)
{
    __shared__ float lA[2][BM * LDSK];
    __shared__ float lB[2][BN * LDSK];

    const int tid    = threadIdx.x;
    const int lane   = tid & 31;
    const int wave   = tid >> 5;
    const int wm     = wave & 3;      // wave M index: 0..3 (32 rows each)
    const int wn     = wave >> 2;     // wave N index: 0..1 (64 cols each)
    const int half   = lane >> 4;     // half-wave select (K-pair / row+8)
    const int l15    = lane & 15;
    const int blockM = blockIdx.y * BM;
    const int blockN = blockIdx.x * BN;

    v8f acc[2][4];
#pragma unroll
    for (int mt = 0; mt < 2; ++mt)
#pragma unroll
        for (int nt = 0; nt < 4; ++nt)
            acc[mt][nt] = {};

    const int nk = K / BK;

#if HAVE_ASYNC_LDS
    issue_tile(A, B, lA[0], lB[0], blockM, blockN, Nv, K, 0, tid);
#endif

    for (int t = 0; t < nk; ++t) {
        const int cur = t & 1;
        float* curA = lA[cur];
        float* curB = lB[cur];

#if HAVE_ASYNC_LDS
        // Prefetch tile t+1 into the other buffer, then drain tile t's 8 ops.
        if (t + 1 < nk) {
            issue_tile(A, B, lA[cur ^ 1], lB[cur ^ 1],
                       blockM, blockN, Nv, K, (t + 1) * BK, tid);
            WAIT_ASYNC(8);
        } else {
            WAIT_ASYNC(0);
        }
#else
        // Fallback: synchronous staging through VGPRs.
        const int kt = t * BK;
#pragma unroll
        for (int i = 0; i < 4; ++i) {
            int slot = tid + i * 256;
            int r    = slot >> 3;
            int c4   = slot & 7;
            const float4 va =
                *(const float4*)(A + (size_t)(blockM + r) * K + kt + c4 * 4);
            float* dst = &curA[r * LDSK + c4 * 4];
            dst[0] = va.x; dst[1] = va.y; dst[2] = va.z; dst[3] = va.w;
        }
#pragma unroll
        for (int i = 0; i < 4; ++i) {
            int slot = tid + i * 256;
            int r    = slot >> 3;
            int c4   = slot & 7;
            int nb   = blockN + r;
            if (nb >= Nv) nb = Nv - 1;
            const float4 vb =
                *(const float4*)(B + (size_t)nb * K + kt + c4 * 4);
            float* dst = &curB[r * LDSK + c4 * 4];
            dst[0] = vb.x; dst[1] = vb.y; dst[2] = vb.z; dst[3] = vb.w;
        }
#endif
        __syncthreads();   // all waves' tile-t data resident in LDS

        // ---- compute: 8 K-steps of 4, 8 WMMAs each -------------------------
#pragma unroll
        for (int k0 = 0; k0 < BK; k0 += 4) {
            // A fragment 16x4 f32 (ISA layout): VGPR0 = K=0 (lanes0-15) /
            // K=2 (lanes16-31); VGPR1 = K=1 / K=3  -> per-lane v2f at column
            // k0 + 2*half, row = tile_row + (lane&15).
            v2f af[2];
#pragma unroll
            for (int mt = 0; mt < 2; ++mt) {
                const float* p =
                    &curA[(wm * 32 + mt * 16 + l15) * LDSK + k0 + half * 2];
                af[mt] = *(const v2f*)p;
            }
            // B fragment 4x16 f32, symmetric layout (row index = N column).
            v2f bf[4];
#pragma unroll
            for (int nt = 0; nt < 4; ++nt) {
                const float* p =
                    &curB[(wn * 64 + nt * 16 + l15) * LDSK + k0 + half * 2];
                bf[nt] = *(const v2f*)p;
            }
#pragma unroll
            for (int mt = 0; mt < 2; ++mt)
#pragma unroll
                for (int nt = 0; nt < 4; ++nt)
                    acc[mt][nt] = __builtin_amdgcn_wmma_f32_16x16x4_f32(
                        /*neg_a=*/false, af[mt], /*neg_b=*/false, bf[nt],
                        /*c_mod=*/(short)0, acc[mt][nt],
                        /*reuse_a=*/false, /*reuse_b=*/false);
        }
        __syncthreads();   // tile t fully consumed before buffer reuse
    }

    // ---- epilogue: mask rows by target bucket, streaming NT stores ---------
    // C/D layout: VGPR r -> (row = tile_row + r + 8*half, col = tile_col + lane&15)
    const int wrow0 = blockM + wm * 32;
#pragma unroll
    for (int mt = 0; mt < 2; ++mt) {
#pragma unroll
        for (int nt = 0; nt < 4; ++nt) {
            int col = blockN + wn * 64 + nt * 16 + l15;
            if (col < Nv) {
#pragma unroll
                for (int r = 0; r < 8; ++r) {
                    int grow = wrow0 + mt * 16 + half * 8 + r;
                    int tg   = target[grow];
                    float s  = (tg >= lo && tg < hi) ? 1.0f : 0.0f;
                    __builtin_nontemporal_store(
                        acc[mt][nt][r] * s,
                        C + (size_t)grow * ldc + col_off + col);
                }
            }
        }
    }
}

// new_head_target: remap in-bucket tokens to cluster tokens, emit as float.
__global__ void adasoft_remap_target(const int* __restrict__ target,
                                     float* __restrict__ out, int n)
{
    int i = blockIdx.x * blockDim.x + threadIdx.x;
    if (i < n) {
        int t  = target[i];
        int nt = t;
        if (t >= 10000 && t < 40000)      nt = 10000;
        else if (t >= 40000 && t < 50257) nt = 10001;
        out[i] = (float)nt;
    }
}

extern "C" void kernel_launch(void* const* d_in, const int* in_sizes, int n_in,
                              void* d_out, int out_size, void* d_ws, size_t ws_size,
                              hipStream_t stream)
{
    const float* x      = (const float*)d_in[0];   // [4096, 1024]
    const int*   target = (const int*)d_in[1];     // [4096]
    const float* head_w = (const float*)d_in[2];   // [10002, 1024]
    const float* t0_w1  = (const float*)d_in[3];   // [256, 1024]
    const float* t0_w2  = (const float*)d_in[4];   // [30000, 256]
    const float* t1_w1  = (const float*)d_in[5];   // [64, 1024]
    const float* t1_w2  = (const float*)d_in[6];   // [10257, 64]
    float* out = (float*)d_out;

    const int N = 4096, D = 1024, LDC = 50259;

    float* H0 = (float*)d_ws;                 // [4096, 256]
    float* H1 = H0 + (size_t)N * 256;         // [4096, 64]

    dim3 blk(256);
    auto grid = [](int nv) { return dim3((nv + BN - 1) / BN, 4096 / BM); };

    // head: columns [0, 10002), no mask
    adasoft_wmma_gemm_nt<<<grid(10002), blk, 0, stream>>>(
        x, head_w, out, target, N, 10002, D, LDC, 0, I32_MIN, I32_MAX);

    // tail hidden activations (no mask): H0 = x @ t0_w1^T, H1 = x @ t1_w1^T
    adasoft_wmma_gemm_nt<<<grid(256), blk, 0, stream>>>(
        x, t0_w1, H0, target, N, 256, D, 256, 0, I32_MIN, I32_MAX);
    adasoft_wmma_gemm_nt<<<grid(64), blk, 0, stream>>>(
        x, t1_w1, H1, target, N, 64, D, 64, 0, I32_MIN, I32_MAX);

    // tail outputs with bucket masks, fused into concat layout
    adasoft_wmma_gemm_nt<<<grid(30000), blk, 0, stream>>>(
        H0, t0_w2, out, target, N, 30000, 256, LDC, 10002, 10000, 40000);
    adasoft_wmma_gemm_nt<<<grid(10257), blk, 0, stream>>>(
        H1, t1_w2, out, target, N, 10257, 64, LDC, 40002, 40000, 50257);

    // second tuple element: remapped head target
    adasoft_remap_target<<<(N + 255) / 256, 256, 0, stream>>>(
        target, out + (size_t)N * LDC, N);
}